// CrossDomainInterestLoss_47768626266727
// MI455X (gfx1250) — compile-verified
//
#include <hip/hip_runtime.h>
#include <hip/hip_bf16.h>
#include <math.h>

#define BSZ 4096
#define DIM 128
#define TILE 64
#define LDS_STRIDE (DIM + 4)   // 132 floats -> 528B rows (16B aligned); frag bank = 4m+kk, conflict-free

#define TAU_INV 20.0f
#define MARGIN 0.3f
#define HARD_NEG_WEIGHT 0.5f

typedef float v2f __attribute__((ext_vector_type(2)));
typedef float v8f __attribute__((ext_vector_type(8)));

// ---------------------------------------------------------------------------
// Zero the per-row accumulators (6 * BSZ floats).
// ---------------------------------------------------------------------------
__global__ __launch_bounds__(256) void zero_kernel(float* __restrict__ p, int n) {
    int i = blockIdx.x * 256 + threadIdx.x;
    if (i < n) p[i] = 0.0f;
}

// ---------------------------------------------------------------------------
// L2-normalize rows of u, a, b into contiguous workspace (3*BSZ rows of DIM).
// One wave per row; each lane loads one float4 (32*4 = 128 elems).
// ---------------------------------------------------------------------------
__global__ __launch_bounds__(256) void normalize_kernel(
    const float* __restrict__ u, const float* __restrict__ a,
    const float* __restrict__ b, float* __restrict__ outBase) {
    int row  = blockIdx.x * 8 + (threadIdx.x >> 5);
    int lane = threadIdx.x & 31;

    const float* src;
    if (row < BSZ)            src = u + (size_t)row * DIM;
    else if (row < 2 * BSZ)   src = a + (size_t)(row - BSZ) * DIM;
    else                      src = b + (size_t)(row - 2 * BSZ) * DIM;

    float4 x = ((const float4*)src)[lane];
    float s = x.x * x.x + x.y * x.y + x.z * x.z + x.w * x.w;
    #pragma unroll
    for (int m = 1; m < 32; m <<= 1) s += __shfl_xor(s, m, 32);

    float scale = 1.0f / fmaxf(sqrtf(s), 1e-12f);
    float4 y; y.x = x.x * scale; y.y = x.y * scale; y.z = x.z * scale; y.w = x.w * scale;
    ((float4*)(outBase + (size_t)row * DIM))[lane] = y;
}

// ---------------------------------------------------------------------------
// Fused sim-tile kernel: one 64x64 tile of sim = U * V^T per block via
// V_WMMA_F32_16X16X4_F32 with 2x2 register blocking (4 WMMAs per 4 LDS frag
// loads), immediately reduced into per-row accumulators:
//   sumExp(row)  += sum_col exp(20*sim)
//   hardSum(row) += sum_col (offdiag & sim>margin) sim
//   hardCnt(row) += count of such cols
// grid = (BSZ/TILE, BSZ/TILE, 2); z=0 -> V=aN/accA, z=1 -> V=bN/accB.
// Block = 128 threads = 4 waves; wave w owns quadrant rows (w&1)*32, cols (w>>1)*32.
// ---------------------------------------------------------------------------
__global__ __launch_bounds__(128) void sim_tile_kernel(
    const float* __restrict__ uN, const float* __restrict__ aN,
    const float* __restrict__ bN, float* __restrict__ accA,
    float* __restrict__ accB) {
    __shared__ float Us[TILE][LDS_STRIDE];
    __shared__ float Vs[TILE][LDS_STRIDE];

    const float* __restrict__ V  = (blockIdx.z == 0) ? aN : bN;
    float* __restrict__ acc      = (blockIdx.z == 0) ? accA : accB;
    float* __restrict__ sumExp   = acc;
    float* __restrict__ hardSum  = acc + BSZ;
    float* __restrict__ hardCnt  = acc + 2 * BSZ;

    const int rowBase = blockIdx.x * TILE;
    const int colBase = blockIdx.y * TILE;
    const int tid = threadIdx.x;

    // Cooperative load: 64 rows x 128 cols as float4 (2048 float4 per matrix).
    #pragma unroll
    for (int i = tid; i < TILE * (DIM / 4); i += 128) {
        int r  = i >> 5;        // row in tile
        int c4 = i & 31;        // float4 index in row
        float4 xu = ((const float4*)(uN + (size_t)(rowBase + r) * DIM))[c4];
        *(float4*)&Us[r][c4 * 4] = xu;
        float4 xv = ((const float4*)(V + (size_t)(colBase + r) * DIM))[c4];
        *(float4*)&Vs[r][c4 * 4] = xv;
    }
    __syncthreads();

    const int wave = tid >> 5;
    const int lane = tid & 31;
    const int half = lane >> 4;        // 0: lanes 0-15, 1: lanes 16-31
    const int l16  = lane & 15;

    const int waveRow = wave & 1;      // 32-row quadrant
    const int waveCol = wave >> 1;     // 32-col quadrant

    // A frag rows (M = lane%16 per half), two row subtiles 16 apart
    const int mr0 = waveRow * 32 + l16;
    const int mr1 = mr0 + 16;
    // B frag rows (N = lane%16 per half), two col subtiles 16 apart
    const int nc0 = waveCol * 32 + l16;
    const int nc1 = nc0 + 16;
    const int kofs = 2 * half;         // lanes 0-15: K={0,1}; lanes 16-31: K={2,3}

    v8f c00 = {0.f, 0.f, 0.f, 0.f, 0.f, 0.f, 0.f, 0.f};
    v8f c01 = c00, c10 = c00, c11 = c00;

    #pragma unroll 4
    for (int k = 0; k < DIM; k += 4) {
        int kk = k + kofs;
        v2f a0 = *(const v2f*)&Us[mr0][kk];
        v2f a1 = *(const v2f*)&Us[mr1][kk];
        v2f b0 = *(const v2f*)&Vs[nc0][kk];
        v2f b1 = *(const v2f*)&Vs[nc1][kk];
        // 8 args: (neg_a, A, neg_b, B, c_mod, C, reuse_a, reuse_b)
        c00 = __builtin_amdgcn_wmma_f32_16x16x4_f32(false, a0, false, b0,
                                                    (short)0, c00, false, false);
        c01 = __builtin_amdgcn_wmma_f32_16x16x4_f32(false, a0, false, b1,
                                                    (short)0, c01, false, false);
        c10 = __builtin_amdgcn_wmma_f32_16x16x4_f32(false, a1, false, b0,
                                                    (short)0, c10, false, false);
        c11 = __builtin_amdgcn_wmma_f32_16x16x4_f32(false, a1, false, b1,
                                                    (short)0, c11, false, false);
    }

    // C/D layout: VGPR j holds (M=j, N=lane) lanes 0-15 and (M=j+8, N=lane-16)
    // lanes 16-31.  Reduce across the 16 columns each lane-half holds.
    const int gcol0 = colBase + waveCol * 32 + l16;
    const int gcol1 = gcol0 + 16;

    #pragma unroll
    for (int j = 0; j < 8; ++j) {
        int grow0 = rowBase + waveRow * 32 + j + 8 * half;
        int grow1 = grow0 + 16;

        float s00 = c00[j], s01 = c01[j];   // row grow0, cols gcol0/gcol1
        float s10 = c10[j], s11 = c11[j];   // row grow1, cols gcol0/gcol1

        float e0 = __expf(TAU_INV * s00) + __expf(TAU_INV * s01);
        float e1 = __expf(TAU_INV * s10) + __expf(TAU_INV * s11);

        float h0 = 0.f, n0 = 0.f, h1 = 0.f, n1 = 0.f;
        if (s00 > MARGIN && grow0 != gcol0) { h0 += s00; n0 += 1.f; }
        if (s01 > MARGIN && grow0 != gcol1) { h0 += s01; n0 += 1.f; }
        if (s10 > MARGIN && grow1 != gcol0) { h1 += s10; n1 += 1.f; }
        if (s11 > MARGIN && grow1 != gcol1) { h1 += s11; n1 += 1.f; }

        // xor masks 1,2,4,8 stay within each 16-lane half
        #pragma unroll
        for (int m = 1; m < 16; m <<= 1) {
            e0 += __shfl_xor(e0, m, 32);
            e1 += __shfl_xor(e1, m, 32);
            h0 += __shfl_xor(h0, m, 32);
            h1 += __shfl_xor(h1, m, 32);
            n0 += __shfl_xor(n0, m, 32);
            n1 += __shfl_xor(n1, m, 32);
        }

        if (l16 == 0) {
            atomicAdd(&sumExp[grow0], e0);
            atomicAdd(&sumExp[grow1], e1);
            if (n0 > 0.f) {
                atomicAdd(&hardSum[grow0], h0);
                atomicAdd(&hardCnt[grow0], n0);
            }
            if (n1 > 0.f) {
                atomicAdd(&hardSum[grow1], h1);
                atomicAdd(&hardCnt[grow1], n1);
            }
        }
    }
}

// ---------------------------------------------------------------------------
// Final scalar reduction over the 4096 per-row accumulators of both matrices.
// ---------------------------------------------------------------------------
__global__ __launch_bounds__(256) void finalize_kernel(
    const float* __restrict__ accA, const float* __restrict__ accB,
    float* __restrict__ out) {
    const int tid = threadIdx.x;
    float lsum = 0.f, hA = 0.f, cA = 0.f, hB = 0.f, cB = 0.f;

    for (int r = tid; r < BSZ; r += 256) {
        float pA = accA[r];
        float pB = accB[r];
        float denom = fmaxf(pA + pB, 1e-9f);
        float ld = logf(denom);
        lsum += (ld - logf(pA)) + (ld - logf(pB));   // lossA + lossB terms

        if (accA[2 * BSZ + r] > 0.f) { hA += accA[BSZ + r]; cA += 1.f; }
        if (accB[2 * BSZ + r] > 0.f) { hB += accB[BSZ + r]; cB += 1.f; }
    }

    __shared__ float red[5 * 256];
    red[0 * 256 + tid] = lsum;
    red[1 * 256 + tid] = hA;
    red[2 * 256 + tid] = cA;
    red[3 * 256 + tid] = hB;
    red[4 * 256 + tid] = cB;
    __syncthreads();
    for (int s = 128; s > 0; s >>= 1) {
        if (tid < s) {
            #pragma unroll
            for (int q = 0; q < 5; ++q)
                red[q * 256 + tid] += red[q * 256 + tid + s];
        }
        __syncthreads();
    }

    if (tid == 0) {
        float base = (red[0 * 256] / (float)BSZ) * 0.5f;  // (lossA+lossB)/2 means
        float tA = red[1 * 256], nA = red[2 * 256];
        float tB = red[3 * 256], nB = red[4 * 256];
        float hLA = (nA > 0.f) ? tA / nA : 0.f;
        float hLB = (nB > 0.f) ? tB / nB : 0.f;
        float weighted = 0.5f * hLA + 1.0f * hLB;
        float total = base + ((fabsf(weighted) > 1e-9f) ? HARD_NEG_WEIGHT * weighted : 0.f);
        out[0] = total;
    }
}

// ---------------------------------------------------------------------------
extern "C" void kernel_launch(void* const* d_in, const int* in_sizes, int n_in,
                              void* d_out, int out_size, void* d_ws, size_t ws_size,
                              hipStream_t stream) {
    const float* u = (const float*)d_in[0];
    const float* a = (const float*)d_in[1];
    const float* b = (const float*)d_in[2];

    float* ws   = (float*)d_ws;
    float* uN   = ws;                          // BSZ*DIM
    float* aN   = uN + (size_t)BSZ * DIM;      // BSZ*DIM
    float* bN   = aN + (size_t)BSZ * DIM;      // BSZ*DIM
    float* accA = bN + (size_t)BSZ * DIM;      // 3*BSZ  (sumExp | hardSum | hardCnt)
    float* accB = accA + 3 * BSZ;              // 3*BSZ

    zero_kernel<<<(6 * BSZ + 255) / 256, 256, 0, stream>>>(accA, 6 * BSZ);
    normalize_kernel<<<(3 * BSZ) / 8, 256, 0, stream>>>(u, a, b, uN);

    dim3 grid(BSZ / TILE, BSZ / TILE, 2);
    sim_tile_kernel<<<grid, 128, 0, stream>>>(uN, aN, bN, accA, accB);

    finalize_kernel<<<1, 256, 0, stream>>>(accA, accB, (float*)d_out);
}